// TemporallyAdaptiveLinearMoE_42039139893874
// MI455X (gfx1250) — compile-verified
//
#include <hip/hip_runtime.h>
#include <hip/hip_bf16.h>

typedef __attribute__((ext_vector_type(2))) float v2f;
typedef __attribute__((ext_vector_type(8))) float v8f;

#define Tt 8
#define Nn 197
#define Cc 384
#define CO 1152
#define NE 4
#define Bb 32
#define BT 256
#define KK 3
#define ES (CO * Cc * KK)   // expert stride in moe_w

// ---------------------------------------------------------------------------
// Kernel A: rf[b][e] = mean_t(cls) . rf_w[e] + rf_b[e]     (32 x 4)
// ---------------------------------------------------------------------------
__global__ void talm_rf_kernel(const float* __restrict__ x,
                               const float* __restrict__ rf_w,
                               const float* __restrict__ rf_b,
                               float* __restrict__ rf_out) {
    const int b = blockIdx.x;
    const int tid = threadIdx.x;          // 128 threads
    __shared__ float red[NE][128];

    float part[NE] = {0.f, 0.f, 0.f, 0.f};
    for (int c = tid; c < Cc; c += 128) {
        float s = 0.f;
        #pragma unroll
        for (int t = 0; t < Tt; ++t)
            s += x[((size_t)(b * Tt + t) * Nn) * Cc + c];
        s *= 0.125f;
        #pragma unroll
        for (int e = 0; e < NE; ++e)
            part[e] = fmaf(s, rf_w[e * Cc + c], part[e]);
    }
    #pragma unroll
    for (int e = 0; e < NE; ++e) red[e][tid] = part[e];
    __syncthreads();
    for (int off = 64; off > 0; off >>= 1) {
        if (tid < off) {
            #pragma unroll
            for (int e = 0; e < NE; ++e) red[e][tid] += red[e][tid + off];
        }
        __syncthreads();
    }
    if (tid < NE) rf_out[b * NE + tid] = red[tid][0] + rf_b[tid];
}

// ---------------------------------------------------------------------------
// Kernel B: routing r[bt][o]  (stored as r[(b*T+t)*1152 + o])
// ---------------------------------------------------------------------------
__global__ void talm_routing_kernel(const float* __restrict__ x,
                                    const float* __restrict__ moe_w,
                                    const float* __restrict__ moe_b,
                                    const float* __restrict__ rf,
                                    float* __restrict__ rout) {
    const int ochunk = blockIdx.x;        // 9 chunks of 128
    const int b      = blockIdx.y;        // 32
    const int tid    = threadIdx.x;       // 128 threads

    __shared__ float clsp[Cc][Tt + 2];    // time-padded cls stream
    __shared__ float rfs[NE];

    if (tid < NE) rfs[tid] = rf[b * NE + tid];
    for (int idx = tid; idx < Cc * Tt; idx += 128) {
        const int t = idx / Cc;
        const int c = idx % Cc;
        clsp[c][t + 1] = x[((size_t)(b * Tt + t) * Nn) * Cc + c];
    }
    for (int c = tid; c < Cc; c += 128) { clsp[c][0] = 0.f; clsp[c][Tt + 1] = 0.f; }
    __syncthreads();

    const int o = ochunk * 128 + tid;     // < 1152 always
    float acc[Tt] = {0.f, 0.f, 0.f, 0.f, 0.f, 0.f, 0.f, 0.f};
    const float r0 = rfs[0], r1 = rfs[1], r2 = rfs[2], r3 = rfs[3];

    for (int c = 0; c < Cc; ++c) {
        const size_t base = ((size_t)o * Cc + c) * KK;
        #pragma unroll
        for (int k = 0; k < KK; ++k) {
            float wm = r0 * moe_w[base + k]
                     + r1 * moe_w[base + (size_t)ES + k]
                     + r2 * moe_w[base + (size_t)2 * ES + k]
                     + r3 * moe_w[base + (size_t)3 * ES + k];
            #pragma unroll
            for (int t = 0; t < Tt; ++t)
                acc[t] = fmaf(wm, clsp[c][t + k], acc[t]);
        }
    }
    const float br = r0 * moe_b[o] + r1 * moe_b[CO + o]
                   + r2 * moe_b[2 * CO + o] + r3 * moe_b[3 * CO + o];
    #pragma unroll
    for (int t = 0; t < Tt; ++t)
        rout[(size_t)(b * Tt + t) * CO + o] = acc[t] + br + 1.0f;
}

// ---------------------------------------------------------------------------
// Kernel C: per-sample GEMM with fp32 WMMA.
//   out[bt, O, n] = sum_c (weight[O,c] * r[bt, seg*384 + c]) * x[bt, n, c]
//                 + bias[O] * r[bt, O]
// Block: 128 threads = 4 waves (2M x 2N). Block tile 128x128; wave tile 64x64
// (4x4 wmma tiles -> 16 wmma per K=4 step from 8 single-b64 fragment loads).
// LDS layout: K-paired float2, row = k>>1 (stride 144), col = m ^ 4*(k>>2).
//   - fragment = one ds_load_b64, lanes 0-15 / 16-31 on disjoint bank halves
//   - transposed b64 stores hit all 64 banks exactly once (XOR swizzle)
// Double-buffered; global side uses b128 loads.
// ---------------------------------------------------------------------------
#define KC 32
#define S2 144   // LDS row stride in float2 units (>=128, ==16 mod 32)

__global__ __launch_bounds__(128)
void talm_main_wmma_kernel(const float* __restrict__ x,
                           const float* __restrict__ weight,
                           const float* __restrict__ bias,
                           const float* __restrict__ rout,
                           float* __restrict__ out) {
    const int bt    = blockIdx.z;                 // 256
    const int Mbase = blockIdx.y * 128;           // 9 tiles -> O
    const int Nbase = blockIdx.x * 128;           // 2 tiles -> n
    const int seg   = Mbase / Cc;                 // 0,1,2 (tile never straddles)

    const int tid    = threadIdx.x;
    const int lane   = tid & 31;
    const int waveId = tid >> 5;                  // 0..3
    const int waveM  = waveId & 1;                // 2 waves over M
    const int waveN  = waveId >> 1;               // 2 waves over N

    __shared__ v2f As2[2][KC / 2][S2];
    __shared__ v2f Bs2[2][KC / 2][S2];
    __shared__ float Rs[Cc];                      // per-sample column scale

    const float* xr = x + (size_t)bt * Nn * Cc;
    const float* rr = rout + (size_t)bt * CO;

    for (int c = tid; c < Cc; c += 128) Rs[c] = rr[seg * Cc + c];
    __syncthreads();

    // G2L thread mapping: a = k-quad index (lane&7), rsub = lane>>3.
    const int aq   = lane & 7;                    // k-quad 0..7
    const int k4   = 4 * aq;
    const int rsub = lane >> 3;                   // 0..3
    const int csw  = 4 * aq;                      // store-side XOR swizzle

    auto loadA = [&](int buf, int k0) {
        const float4 rsv = *(const float4*)&Rs[k0 + k4];
        #pragma unroll
        for (int i = 0; i < 8; ++i) {
            const int m = waveId * 32 + i * 4 + rsub;         // covers 0..127
            const float4 v =
                *(const float4*)&weight[(size_t)(Mbase + m) * Cc + k0 + k4];
            const int colS = m ^ csw;
            As2[buf][2 * aq + 0][colS] = (v2f){v.x * rsv.x, v.y * rsv.y};
            As2[buf][2 * aq + 1][colS] = (v2f){v.z * rsv.z, v.w * rsv.w};
        }
    };
    auto loadB = [&](int buf, int k0) {
        #pragma unroll
        for (int i = 0; i < 8; ++i) {
            const int nl = waveId * 32 + i * 4 + rsub;        // covers 0..127
            const int n  = Nbase + nl;
            float4 v = make_float4(0.f, 0.f, 0.f, 0.f);
            if (n < Nn) v = *(const float4*)&xr[(size_t)n * Cc + k0 + k4];
            const int colS = nl ^ csw;
            Bs2[buf][2 * aq + 0][colS] = (v2f){v.x, v.y};
            Bs2[buf][2 * aq + 1][colS] = (v2f){v.z, v.w};
        }
    };

    v8f acc[4][4] = {};

    const int mA   = waveM * 64 + (lane & 15);    // + 16*fi
    const int nB   = waveN * 64 + (lane & 15);    // + 16*fj
    const int rowH = lane >> 4;                   // 0 or 1 (K-pair row offset)

    loadA(0, 0);
    loadB(0, 0);

    const int nsteps = Cc / KC;                   // 12
    for (int step = 0; step < nsteps; ++step) {
        __syncthreads();
        const int cur = step & 1;
        if (step + 1 < nsteps) {
            loadA(cur ^ 1, (step + 1) * KC);
            loadB(cur ^ 1, (step + 1) * KC);
        }
        #pragma unroll
        for (int kk = 0; kk < KC; kk += 4) {
            const int row = (kk >> 1) + rowH;     // (kk + 2*(lane>>4)) / 2
            const int sw  = 4 * (kk >> 2);        // compile-time constant
            v2f a[4], b[4];
            #pragma unroll
            for (int f = 0; f < 4; ++f) {
                a[f] = As2[cur][row][(mA + 16 * f) ^ sw];
                b[f] = Bs2[cur][row][(nB + 16 * f) ^ sw];
            }
            #pragma unroll
            for (int mi = 0; mi < 4; ++mi)
                #pragma unroll
                for (int ni = 0; ni < 4; ++ni)
                    acc[mi][ni] = __builtin_amdgcn_wmma_f32_16x16x4_f32(
                        false, a[mi], false, b[ni], (short)0, acc[mi][ni],
                        false, false);
        }
    }

    // ---- epilogue: + bias[O]*r[bt,O], masked store -------------------------
    const int nlo  = lane & 15;
    const int mhi8 = 8 * (lane >> 4);
    float* outp = out + (size_t)bt * CO * Nn;

    #pragma unroll
    for (int mi = 0; mi < 4; ++mi) {
        #pragma unroll
        for (int ni = 0; ni < 4; ++ni) {
            const int n = Nbase + waveN * 64 + 16 * ni + nlo;
            if (n >= Nn) continue;
            const int Mf = Mbase + waveM * 64 + 16 * mi;
            #pragma unroll
            for (int v = 0; v < 8; ++v) {
                const int O = Mf + v + mhi8;
                outp[(size_t)O * Nn + n] = acc[mi][ni][v] + bias[O] * rr[O];
            }
        }
    }
}

// ---------------------------------------------------------------------------
extern "C" void kernel_launch(void* const* d_in, const int* in_sizes, int n_in,
                              void* d_out, int out_size, void* d_ws, size_t ws_size,
                              hipStream_t stream) {
    const float* x      = (const float*)d_in[0];
    const float* rf_w   = (const float*)d_in[1];
    const float* rf_b   = (const float*)d_in[2];
    const float* moe_w  = (const float*)d_in[3];
    const float* moe_b  = (const float*)d_in[4];
    const float* weight = (const float*)d_in[5];
    const float* bias   = (const float*)d_in[6];
    float* out = (float*)d_out;

    float* wsf  = (float*)d_ws;
    float* rf   = wsf;                 // 32*4
    float* rout = wsf + 128;           // 256*1152

    talm_rf_kernel<<<dim3(Bb), dim3(128), 0, stream>>>(x, rf_w, rf_b, rf);
    talm_routing_kernel<<<dim3(9, Bb), dim3(128), 0, stream>>>(x, moe_w, moe_b, rf, rout);
    talm_main_wmma_kernel<<<dim3(2, 9, BT), dim3(128), 0, stream>>>(x, weight, bias, rout, out);
}